// CosmicFlashAttention_70600672411787
// MI455X (gfx1250) — compile-verified
//
#include <hip/hip_runtime.h>

typedef __attribute__((ext_vector_type(2))) float v2f;
typedef __attribute__((ext_vector_type(8))) float v8f;

#define NUM_HEADS 16
#define KEY_SIZE  64
#define TOKEN_F   (NUM_HEADS * KEY_SIZE)  // 1024 floats per token per tensor
#define QS 68   // Q/K tile row stride in floats (4*row + k banks -> conflict free)
#define VS 72   // V tile row stride (8*k + col banks -> halves disjoint)
#define WS 20   // weight tile row stride
#define WAVES_PER_BLOCK 4

// ds_swizzle group-of-32 XOR butterfly with immediate mask:
// offset = (xor_mask << 10) | (or_mask << 5) | and_mask(0x1f)
template <int IMM>
__device__ __forceinline__ float swz_xor(float x) {
  return __int_as_float(__builtin_amdgcn_ds_swizzle(__float_as_int(x), IMM));
}

__global__ __launch_bounds__(WAVES_PER_BLOCK * 32)
void headmix_attn_wmma(const float* __restrict__ q,
                       const float* __restrict__ k,
                       const float* __restrict__ v,
                       float* __restrict__ out,
                       int nTokens) {
  // per-wave private LDS region: Q tile, K tile, V tile, weight tile
  __shared__ __align__(16) float lds[WAVES_PER_BLOCK * (16*QS + 16*QS + 16*VS + 16*WS)];

  const int wave = threadIdx.x >> 5;
  const int lane = threadIdx.x & 31;
  const int half = lane >> 4;   // 0: lanes 0-15, 1: lanes 16-31
  const int l16  = lane & 15;

  float* qT = lds + wave * (16*QS + 16*QS + 16*VS + 16*WS);
  float* kT = qT + 16*QS;
  float* vT = kT + 16*QS;
  float* wT = vT + 16*VS;

  const int token = blockIdx.x * WAVES_PER_BLOCK + wave;
  if (token >= nTokens) return;  // wave-uniform: EXEC stays all-ones for WMMA
  const long long base = (long long)token * TOKEN_F;

  // ---- coalesced global -> LDS staging (512B per wave-load) ----
#pragma unroll
  for (int i = 0; i < 8; ++i) {
    const int fi  = (i * 32 + lane) * 4;   // float index within 1024-float tile
    const int row = fi >> 6;               // head
    const int col = fi & 63;               // key_size element (multiple of 4)
    *(float4*)(qT + row * QS + col) = *(const float4*)(q + base + fi);
    *(float4*)(kT + row * QS + col) = *(const float4*)(k + base + fi);
    *(float4*)(vT + row * VS + col) = *(const float4*)(v + base + fi);
  }

  // ---- logits = Q @ K^T : 16 chained V_WMMA_F32_16X16X4_F32 over K=64 ----
  v8f c = {0.f, 0.f, 0.f, 0.f, 0.f, 0.f, 0.f, 0.f};
#pragma unroll
  for (int ks = 0; ks < KEY_SIZE; ks += 4) {
    const int kk = ks + half * 2;
    // A: lane holds Q[M=l16][kk], Q[M=l16][kk+1]
    v2f a = *(const v2f*)(qT + l16 * QS + kk);
    // B = K^T: lane holds K[N=l16][kk], K[N=l16][kk+1]
    v2f b = *(const v2f*)(kT + l16 * QS + kk);
    c = __builtin_amdgcn_wmma_f32_16x16x4_f32(false, a, false, b,
                                              (short)0, c, false, false);
  }

  // ---- softmax over N (key-head axis): 16-lane XOR reductions via ds_swizzle ----
  const float scale = 0.125f;  // 1/sqrt(64)
#pragma unroll
  for (int r = 0; r < 8; ++r) {
    float x = c[r] * scale;
    float m = x;
    m = fmaxf(m, swz_xor<0x041f>(m));   // xor 1
    m = fmaxf(m, swz_xor<0x081f>(m));   // xor 2
    m = fmaxf(m, swz_xor<0x101f>(m));   // xor 4
    m = fmaxf(m, swz_xor<0x201f>(m));   // xor 8
    float e = __expf(x - m);
    float s = e;
    s += swz_xor<0x041f>(s);
    s += swz_xor<0x081f>(s);
    s += swz_xor<0x101f>(s);
    s += swz_xor<0x201f>(s);
    // D layout: VGPR r holds row M = r + 8*half, column N = l16
    wT[(r + 8 * half) * WS + l16] = e / s;
  }

  // ---- out = W @ V : per 16-wide N chunk, 4 chained WMMAs over K=16 ----
#pragma unroll
  for (int n = 0; n < KEY_SIZE; n += 16) {
    v8f acc = {0.f, 0.f, 0.f, 0.f, 0.f, 0.f, 0.f, 0.f};
#pragma unroll
    for (int ks = 0; ks < 16; ks += 4) {
      const int kk = ks + half * 2;
      // A: lane holds W[M=l16][kk], W[M=l16][kk+1]
      v2f a = *(const v2f*)(wT + l16 * WS + kk);
      // B: lane holds V[kk][n+l16], V[kk+1][n+l16]
      v2f b;
      b.x = vT[kk * VS + n + l16];
      b.y = vT[(kk + 1) * VS + n + l16];
      acc = __builtin_amdgcn_wmma_f32_16x16x4_f32(false, a, false, b,
                                                  (short)0, acc, false, false);
    }
    // stage output chunk into the (now dead) Q tile for coalesced writeback
#pragma unroll
    for (int r = 0; r < 8; ++r)
      qT[(r + 8 * half) * QS + n + l16] = acc[r];
  }

  // ---- coalesced LDS -> global writeback ----
#pragma unroll
  for (int i = 0; i < 8; ++i) {
    const int fi  = (i * 32 + lane) * 4;
    const int row = fi >> 6;
    const int col = fi & 63;
    *(float4*)(out + base + fi) = *(const float4*)(qT + row * QS + col);
  }
}

extern "C" void kernel_launch(void* const* d_in, const int* in_sizes, int n_in,
                              void* d_out, int out_size, void* d_ws, size_t ws_size,
                              hipStream_t stream) {
  (void)n_in; (void)out_size; (void)d_ws; (void)ws_size;
  const float* q = (const float*)d_in[0];
  const float* k = (const float*)d_in[1];
  const float* v = (const float*)d_in[2];
  float* out = (float*)d_out;

  const int nTokens = in_sizes[0] / TOKEN_F;  // B*S = 32768
  const int blocks  = (nTokens + WAVES_PER_BLOCK - 1) / WAVES_PER_BLOCK;
  headmix_attn_wmma<<<blocks, WAVES_PER_BLOCK * 32, 0, stream>>>(q, k, v, out, nTokens);
}